// PointNet2_40381282517043
// MI455X (gfx1250) — compile-verified
//
#include <hip/hip_runtime.h>
#include <math.h>

// ---------------------------------------------------------------------------
// PointNet2 head for MI455X (gfx1250), wave32, WMMA f32 16x16x4.
// B=2 T=5 N=4096 C=16 D=128 NSAMPLE=4; M=B*N=8192, Ns=16384, MS=M*4=32768.
// d_out = [out (M*7 floats) | feature (M*128 floats)]
// ---------------------------------------------------------------------------

typedef float v2f __attribute__((ext_vector_type(2)));
typedef float v8f __attribute__((ext_vector_type(8)));

#define NQ        4096
#define NSRC      16384
#define M_TOT     8192        // B*N
#define MS_TOT    32768       // M*4
#define DCH       128

// workspace layout (bytes)
#define OFF_IDX     0                         // int32 [MS_TOT]            128 KB
#define OFF_STATS1  131072                    // float [256] (sum|sumsq)   1 KB
#define OFF_STATS2  132096                    // float [3*256]             3 KB
#define OFF_H       135168                    // float [MS_TOT*128]        16 MB
#define OFF_Y       16912384                  // float [3*M_TOT*128]       12 MB

// ---------------------------------------------------------------------------
// 1) Radius ball query: first 4 source indices with d2 < 1.0 (== reference's
//    top_k over -index scores). idx = -1,-1,-1,-1 marks an empty group.
// ---------------------------------------------------------------------------
__global__ __launch_bounds__(256) void ballq_kernel(const float* __restrict__ x,
                                                    int* __restrict__ idx)
{
    int b = blockIdx.y;
    int q = blockIdx.x * 256 + threadIdx.x;

    const float4* qp4 = (const float4*)(x + ((size_t)(b * 5) * NQ + q) * 16);
    float4 qp = *qp4;                                    // .x .y .z = query point
    const float4* src = (const float4*)(x + (size_t)(b * 5 + 1) * NQ * 16);

    int found[4];
    int cnt = 0;
    for (int j = 0; j < NSRC; ++j) {
        float4 p = src[(size_t)j * 4];                   // 64B-strided, aligned
        float dx = qp.x - p.x, dy = qp.y - p.y, dz = qp.z - p.z;
        float d2 = dx * dx + dy * dy + dz * dz;
        if (d2 < 1.0f) {
            found[cnt++] = j;
            if (cnt == 4) break;                         // early exit (~50 iters avg)
        }
    }
    int4 r;
    if (cnt == 0) {
        r = make_int4(-1, -1, -1, -1);
    } else {
        r.x = found[0];
        r.y = (cnt > 1) ? found[1] : found[0];
        r.z = (cnt > 2) ? found[2] : found[0];
        r.w = (cnt > 3) ? found[3] : found[0];
    }
    ((int4*)idx)[(size_t)b * NQ + q] = r;
}

// ---------------------------------------------------------------------------
// 2) Fused gather + conv1 via WMMA f32 16x16x4 (K=16 as 4 chained steps) +
//    per-channel BN statistics (sum, sumsq over all 32768 (m,s) rows).
//    conv1_b skipped: constant per-channel bias cancels exactly in batch-norm.
// ---------------------------------------------------------------------------
__global__ __launch_bounds__(256) void conv1_wmma_kernel(
    const float* __restrict__ x, const float* __restrict__ w,   // w: (128,16)
    const int* __restrict__ idx, float* __restrict__ H,
    float* __restrict__ stats1)
{
    __shared__ float As[16][17];        // 16 (m,s)-rows x 16 channels, padded
    __shared__ float s_sum[DCH], s_sq[DCH];

    int tid = threadIdx.x;
    if (tid < DCH) { s_sum[tid] = 0.f; s_sq[tid] = 0.f; }

    // gather A tile: one element per thread
    int row = tid >> 4, c = tid & 15;
    int r   = blockIdx.x * 16 + row;            // global (m*4+s) row
    int j   = idx[r];
    float a = 0.f;
    if (j >= 0) {
        int m = r >> 2;
        int b = m >> 12;                        // m / 4096
        a = x[(size_t)(b * 5 + 1) * NQ * 16 + (size_t)j * 16 + c];
    }
    As[row][c] = a;
    __syncthreads();

    int wave = tid >> 5, lane = tid & 31;
    int o0 = wave * 16;                         // 8 waves cover all 128 channels
    int Mr = lane & 15, Khalf = lane >> 4;

    v8f acc = {};
#pragma unroll
    for (int kk = 0; kk < 4; ++kk) {
        int kb = kk * 4 + Khalf * 2;
        v2f av; av.x = As[Mr][kb];               av.y = As[Mr][kb + 1];
        v2f bv; bv.x = w[(o0 + Mr) * 16 + kb];   bv.y = w[(o0 + Mr) * 16 + kb + 1];
        acc = __builtin_amdgcn_wmma_f32_16x16x4_f32(
            false, av, false, bv, (short)0, acc, false, false);
    }

    // C/D layout: lane holds col N=lane%16, rows Mbase..Mbase+7 (Mbase=8*(lane/16))
    int Nc = Mr, Mbase = Khalf * 8;
    float s = 0.f, sq = 0.f;
    size_t base = (size_t)blockIdx.x * 16 * DCH + (o0 + Nc);
#pragma unroll
    for (int rr = 0; rr < 8; ++rr) {
        float v = acc[rr];
        H[base + (size_t)(Mbase + rr) * DCH] = v;
        s += v; sq += v * v;
    }
    atomicAdd(&s_sum[o0 + Nc], s);
    atomicAdd(&s_sq[o0 + Nc], sq);
    __syncthreads();
    if (tid < DCH) {
        atomicAdd(&stats1[tid], s_sum[tid]);
        atomicAdd(&stats1[DCH + tid], s_sq[tid]);
    }
}

// ---------------------------------------------------------------------------
// 3) BN (train-mode, stats over (m,s)) + max over s  -> feature (M x 128)
// ---------------------------------------------------------------------------
__global__ __launch_bounds__(256) void bnmax_kernel(
    const float* __restrict__ H, const float* __restrict__ stats1,
    const float* __restrict__ g, const float* __restrict__ bb,
    float* __restrict__ feat)
{
    int t = blockIdx.x * 256 + threadIdx.x;     // over M*128
    int o = t & 127, m = t >> 7;
    const float inv_cnt = 1.0f / (float)MS_TOT;
    float mean = stats1[o] * inv_cnt;
    float var  = stats1[DCH + o] * inv_cnt - mean * mean;
    float scale = g[o] * rsqrtf(var + 1e-5f);
    float shift = bb[o] - mean * scale;

    size_t base = (size_t)m * 4 * DCH + o;
    float mx = -INFINITY;
#pragma unroll
    for (int s = 0; s < 4; ++s)
        mx = fmaxf(mx, scale * H[base + s * DCH] + shift);
    feat[t] = mx;
}

// ---------------------------------------------------------------------------
// 4) fc1/fc2/fc3 (blockIdx.y selects which): Y = feat(8192x128) @ W^T(128x128)
//    via 32 chained WMMA f32 16x16x4, + per-channel BN stats (axis 0).
// ---------------------------------------------------------------------------
__global__ __launch_bounds__(256) void fc_wmma_kernel(
    const float* __restrict__ feat,
    const float* __restrict__ w1, const float* __restrict__ w2,
    const float* __restrict__ w3,
    float* __restrict__ Y, float* __restrict__ stats2)
{
    __shared__ float As[16][132];               // 16 rows x 128, padded stride
    __shared__ float s_sum[DCH], s_sq[DCH];

    int z = blockIdx.y;
    const float* w = (z == 0) ? w1 : ((z == 1) ? w2 : w3);

    int tid = threadIdx.x;
    if (tid < DCH) { s_sum[tid] = 0.f; s_sq[tid] = 0.f; }

    int m0 = blockIdx.x * 16;
    for (int i = tid; i < 16 * DCH; i += 256)
        As[i >> 7][i & 127] = feat[(size_t)(m0 + (i >> 7)) * DCH + (i & 127)];
    __syncthreads();

    int wave = tid >> 5, lane = tid & 31;
    int o0 = wave * 16;
    int Mr = lane & 15, Khalf = lane >> 4;

    v8f acc = {};
#pragma unroll
    for (int kk = 0; kk < 32; ++kk) {
        int kb = kk * 4 + Khalf * 2;
        v2f av; av.x = As[Mr][kb];                av.y = As[Mr][kb + 1];
        v2f bv; bv.x = w[(o0 + Mr) * DCH + kb];   bv.y = w[(o0 + Mr) * DCH + kb + 1];
        acc = __builtin_amdgcn_wmma_f32_16x16x4_f32(
            false, av, false, bv, (short)0, acc, false, false);
    }

    int Nc = Mr, Mbase = Khalf * 8;
    size_t zoff = (size_t)z * M_TOT * DCH;
    float s = 0.f, sq = 0.f;
#pragma unroll
    for (int rr = 0; rr < 8; ++rr) {
        float v = acc[rr];
        Y[zoff + (size_t)(m0 + Mbase + rr) * DCH + (o0 + Nc)] = v;
        s += v; sq += v * v;
    }
    atomicAdd(&s_sum[o0 + Nc], s);
    atomicAdd(&s_sq[o0 + Nc], sq);
    __syncthreads();
    if (tid < DCH) {
        atomicAdd(&stats2[z * 256 + tid], s_sum[tid]);
        atomicAdd(&stats2[z * 256 + DCH + tid], s_sq[tid]);
    }
}

// ---------------------------------------------------------------------------
// 5) heads: relu(bn(Y_z)) @ head_w^T + head_b -> out (M x 7)
// ---------------------------------------------------------------------------
__global__ __launch_bounds__(256) void heads_kernel(
    const float* __restrict__ Y, const float* __restrict__ stats2,
    const float* __restrict__ g, const float* __restrict__ bb,
    const float* __restrict__ wce,  const float* __restrict__ bce,
    const float* __restrict__ wlwh, const float* __restrict__ blwh,
    const float* __restrict__ wth,  const float* __restrict__ bth,
    float* __restrict__ out)
{
    __shared__ float sc[3][DCH], sh[3][DCH];
    int tid = threadIdx.x;
    if (tid < DCH) {
        const float inv_cnt = 1.0f / (float)M_TOT;
        for (int z = 0; z < 3; ++z) {
            float mean = stats2[z * 256 + tid] * inv_cnt;
            float var  = stats2[z * 256 + DCH + tid] * inv_cnt - mean * mean;
            float scale = g[tid] * rsqrtf(var + 1e-5f);
            sc[z][tid] = scale;
            sh[z][tid] = bb[tid] - mean * scale;
        }
    }
    __syncthreads();

    int m = blockIdx.x * 256 + tid;
    float acc[7] = { bce[0], bce[1], bce[2], blwh[0], blwh[1], blwh[2], bth[0] };
    size_t r0 = (size_t)m * DCH;
    for (int ch = 0; ch < DCH; ++ch) {
        float y1 = fmaxf(sc[0][ch] * Y[r0 + ch]                          + sh[0][ch], 0.f);
        float y2 = fmaxf(sc[1][ch] * Y[(size_t)M_TOT * DCH     + r0 + ch] + sh[1][ch], 0.f);
        float y3 = fmaxf(sc[2][ch] * Y[(size_t)2 * M_TOT * DCH + r0 + ch] + sh[2][ch], 0.f);
        acc[0] += y1 * wce[0 * DCH + ch];
        acc[1] += y1 * wce[1 * DCH + ch];
        acc[2] += y1 * wce[2 * DCH + ch];
        acc[3] += y2 * wlwh[0 * DCH + ch];
        acc[4] += y2 * wlwh[1 * DCH + ch];
        acc[5] += y2 * wlwh[2 * DCH + ch];
        acc[6] += y3 * wth[ch];
    }
#pragma unroll
    for (int k = 0; k < 7; ++k) out[(size_t)m * 7 + k] = acc[k];
}

// ---------------------------------------------------------------------------
extern "C" void kernel_launch(void* const* d_in, const int* in_sizes, int n_in,
                              void* d_out, int out_size, void* d_ws, size_t ws_size,
                              hipStream_t stream)
{
    const float* x          = (const float*)d_in[0];
    const float* conv1_w    = (const float*)d_in[1];
    // d_in[2] conv1_b : cancels exactly through batch-norm (skipped)
    const float* bn_g       = (const float*)d_in[3];
    const float* bn_b       = (const float*)d_in[4];
    const float* fc1_w      = (const float*)d_in[5];
    // d_in[6] fc1_b : cancels in BN
    const float* fc_ce_w    = (const float*)d_in[7];
    const float* fc_ce_b    = (const float*)d_in[8];
    const float* fc2_w      = (const float*)d_in[9];
    // d_in[10] fc2_b : cancels in BN
    const float* fc_lwh_w   = (const float*)d_in[11];
    const float* fc_lwh_b   = (const float*)d_in[12];
    const float* fc3_w      = (const float*)d_in[13];
    // d_in[14] fc3_b : cancels in BN
    const float* fc_theta_w = (const float*)d_in[15];
    const float* fc_theta_b = (const float*)d_in[16];

    char*  ws     = (char*)d_ws;
    int*   idx    = (int*)(ws + OFF_IDX);
    float* stats1 = (float*)(ws + OFF_STATS1);
    float* stats2 = (float*)(ws + OFF_STATS2);
    float* H      = (float*)(ws + OFF_H);
    float* Y      = (float*)(ws + OFF_Y);

    float* out  = (float*)d_out;
    float* feat = out + (size_t)M_TOT * 7;      // feature region of d_out

    // zero the BN stats accumulators (stats1 + stats2 are contiguous)
    hipMemsetAsync(ws + OFF_STATS1, 0, 4096, stream);

    ballq_kernel     <<<dim3(NQ / 256, 2), 256, 0, stream>>>(x, idx);
    conv1_wmma_kernel<<<MS_TOT / 16,        256, 0, stream>>>(x, conv1_w, idx, H, stats1);
    bnmax_kernel     <<<M_TOT * DCH / 256,  256, 0, stream>>>(H, stats1, bn_g, bn_b, feat);
    fc_wmma_kernel   <<<dim3(M_TOT / 16, 3),256, 0, stream>>>(feat, fc1_w, fc2_w, fc3_w, Y, stats2);
    heads_kernel     <<<M_TOT / 256,        256, 0, stream>>>(Y, stats2, bn_g, bn_b,
                                                              fc_ce_w, fc_ce_b,
                                                              fc_lwh_w, fc_lwh_b,
                                                              fc_theta_w, fc_theta_b, out);
}